// AdvancedMixConsole_7327214207208
// MI455X (gfx1250) — compile-verified
//
#include <hip/hip_runtime.h>

// ---------------------------------------------------------------------------
// AdvancedMixConsole on MI455X (gfx1250)
//
// All heavy math is FFT convolution. Strategy for this part:
//  * N = 2^19 FFTs use the four-step algorithm N = 4096 x 128:
//      - fft4096_wmma : 3 fused radix-16 Stockham stages in LDS (64 KB
//        ping-pong), butterflies on V_WMMA_F32_16X16X4_F32 (full fp32).
//      - fft128_twiddle : 128-pt FFT (radix 16 WMMA + radix 8 VALU) in LDS,
//        fused with the inter-step twiddle e^(+-2pi i n1 k2 / N).
//    => 2 coalesced global passes instead of 5 (~5 GB vs ~12.5 GB stage
//       traffic @ 23.3 TB/s HBM; this workload is bandwidth-bound).
//    Spectra live in scrambled (k2,k1) layout; per-bin EQ / compressor
//    responses are evaluated on the fly and map idx -> k = k2 + 128*k1.
//  * N = 2^17 IR-build FFTs keep a global mixed-radix Stockham path
//    (radix-16 WMMA stages + radix-2 tail), all power-of-two indexing done
//    with shifts (no runtime integer division sequences).
// ---------------------------------------------------------------------------

#define SRF     44100.0f
#define BSZ     2
#define NTR     16
#define SEQL    262144
#define REVN    65536
#define TAPSN   1023
#define NOISEL  (REVN + TAPSN - 1)      // 66558
#define NFA     ((int)(1u << 19))       // main conv FFT length
#define LOGA    19
#define NFB     ((int)(1u << 17))       // IR build FFT length
#define LOGB    17
#define N1F     4096                    // four-step: big FFT size
#define N2F     128                     // four-step: small FFT size
#define PI_F    3.14159265358979f
#define DB2LIN  0.11512925465f          // ln(10)/20
#define LN10_40 0.05756462732f          // ln(10)/40
#define PANNED_SZ ((size_t)BSZ * 2 * NTR * SEQL)

typedef float v2f __attribute__((ext_vector_type(2)));
typedef float v8f __attribute__((ext_vector_type(8)));

// ------------------------- parameter denorm tables -------------------------
__constant__ float TRACK_R[27][2] = {
    {-48.f, 48.f},
    {-12.f, 12.f}, {20.f, 2000.f},    {0.1f, 5.f},
    {-12.f, 12.f}, {80.f, 2000.f},    {0.1f, 5.f},
    {-12.f, 12.f}, {2000.f, 8000.f},  {0.1f, 5.f},
    {-12.f, 12.f}, {8000.f, 12000.f}, {0.1f, 5.f},
    {-12.f, 12.f}, {12000.f, 21050.f},{0.1f, 5.f},
    {-12.f, 12.f}, {6000.f, 21050.f}, {0.1f, 5.f},
    {-60.f, 0.f}, {1.f, 10.f}, {5.f, 250.f}, {10.f, 250.f}, {3.f, 12.f}, {0.f, 6.f},
    {-80.f, 12.f}, {0.f, 1.f}
};
__constant__ float MASTER_R[26][2] = {
    {-48.f, 48.f},
    {-12.f, 12.f}, {20.f, 2000.f},    {0.1f, 5.f},
    {-12.f, 12.f}, {80.f, 2000.f},    {0.1f, 5.f},
    {-12.f, 12.f}, {2000.f, 8000.f},  {0.1f, 5.f},
    {-12.f, 12.f}, {8000.f, 12000.f}, {0.1f, 5.f},
    {-12.f, 12.f}, {12000.f, 21050.f},{0.1f, 5.f},
    {-12.f, 12.f}, {6000.f, 21050.f}, {0.1f, 5.f},
    {-60.f, 0.f}, {1.f, 10.f}, {5.f, 250.f}, {10.f, 250.f}, {3.f, 12.f}, {0.f, 6.f},
    {-48.f, 48.f}
};

// ------------------------------ complex helpers ----------------------------
__device__ __forceinline__ float2 cmul(float2 a, float2 b) {
    return make_float2(a.x * b.x - a.y * b.y, a.x * b.y + a.y * b.x);
}
__device__ __forceinline__ float2 cdiv(float2 a, float2 b) {
    float d = 1.0f / (b.x * b.x + b.y * b.y + 1e-30f);
    return make_float2((a.x * b.x + a.y * b.y) * d, (a.y * b.x - a.x * b.y) * d);
}

// Biquad response at z1=e^{-iw}; type 0=low_shelf, 1=peaking, 2=high_shelf
__device__ __forceinline__ float2 biquad_h(float gdb, float fc, float q, int type,
                                           float2 z1, float2 z2) {
    float Aa = __expf(gdb * LN10_40);
    float w0 = (2.f * PI_F / SRF) * fc;
    float s0, c0; __sincosf(w0, &s0, &c0);
    float al = s0 / (2.f * q);
    float sA = sqrtf(Aa);
    float b0, b1, b2, a0, a1, a2;
    if (type == 0) {
        b0 = Aa * ((Aa + 1.f) - (Aa - 1.f) * c0 + 2.f * sA * al);
        b1 = 2.f * Aa * ((Aa - 1.f) - (Aa + 1.f) * c0);
        b2 = Aa * ((Aa + 1.f) - (Aa - 1.f) * c0 - 2.f * sA * al);
        a0 = (Aa + 1.f) + (Aa - 1.f) * c0 + 2.f * sA * al;
        a1 = -2.f * ((Aa - 1.f) + (Aa + 1.f) * c0);
        a2 = (Aa + 1.f) + (Aa - 1.f) * c0 - 2.f * sA * al;
    } else if (type == 2) {
        b0 = Aa * ((Aa + 1.f) + (Aa - 1.f) * c0 + 2.f * sA * al);
        b1 = -2.f * Aa * ((Aa - 1.f) + (Aa + 1.f) * c0);
        b2 = Aa * ((Aa + 1.f) + (Aa - 1.f) * c0 - 2.f * sA * al);
        a0 = (Aa + 1.f) - (Aa - 1.f) * c0 + 2.f * sA * al;
        a1 = 2.f * ((Aa - 1.f) - (Aa + 1.f) * c0);
        a2 = (Aa + 1.f) - (Aa - 1.f) * c0 - 2.f * sA * al;
    } else {
        b0 = 1.f + al * Aa; b1 = -2.f * c0; b2 = 1.f - al * Aa;
        a0 = 1.f + al / Aa; a1 = -2.f * c0; a2 = 1.f - al / Aa;
    }
    float2 num = make_float2(b0 + b1 * z1.x + b2 * z2.x, b1 * z1.y + b2 * z2.y);
    float2 den = make_float2(a0 + a1 * z1.x + a2 * z2.x, a1 * z1.y + a2 * z2.y);
    return cdiv(num, den);
}

// --------------------- DFT16 A-matrix fragments (WMMA) ---------------------
// 16x4 f32 A layout: lanes 0-15 row m with K = k0,k0+1; lanes 16-31 K+2.
__device__ __forceinline__ void dft16_frags(int lane, float dirSign, v2f *ar, v2f *ai) {
    const int m = lane & 15;
    const int khalf = (lane >> 4) << 1;
#pragma unroll
    for (int cc = 0; cc < 4; ++cc) {
        int k0 = 4 * cc + khalf;
        float t0 = (PI_F / 8.0f) * (float)((m * k0) & 15);
        float t1 = (PI_F / 8.0f) * (float)((m * (k0 + 1)) & 15);
        float s0, c0, s1, c1;
        __sincosf(t0, &s0, &c0);
        __sincosf(t1, &s1, &c1);
        v2f r; r.x = c0; r.y = c1;
        v2f i; i.x = dirSign * s0; i.y = dirSign * s1;
        ar[cc] = r; ai[cc] = i;
    }
}

// One complex 16x16 DFT tile via 16 chained f32 WMMAs (K accumulated 4 at a time)
#define WMMA_CPLX_TILE(LOADX0, LOADX1)                                                   \
    v8f yr = {0.f, 0.f, 0.f, 0.f, 0.f, 0.f, 0.f, 0.f};                                   \
    v8f yi = {0.f, 0.f, 0.f, 0.f, 0.f, 0.f, 0.f, 0.f};                                   \
    _Pragma("unroll")                                                                    \
    for (int cc = 0; cc < 4; ++cc) {                                                     \
        const int k0 = 4 * cc + khalf;                                                   \
        float2 x0 = (LOADX0);                                                            \
        float2 x1 = (LOADX1);                                                            \
        v2f br; br.x = x0.x; br.y = x1.x;                                                \
        v2f bi; bi.x = x0.y; bi.y = x1.y;                                                \
        v2f nai = -ai[cc];                                                               \
        yr = __builtin_amdgcn_wmma_f32_16x16x4_f32(false, ar[cc], false, br, (short)0, yr, false, false); \
        yr = __builtin_amdgcn_wmma_f32_16x16x4_f32(false, nai,    false, bi, (short)0, yr, false, false); \
        yi = __builtin_amdgcn_wmma_f32_16x16x4_f32(false, ar[cc], false, bi, (short)0, yi, false, false); \
        yi = __builtin_amdgcn_wmma_f32_16x16x4_f32(false, ai[cc], false, br, (short)0, yi, false, false); \
    }

// ===================== four-step FFT: 4096-pt LDS kernel ===================
// One workgroup = one contiguous row of 4096 complex; 3 radix-16 Stockham
// stages fused in LDS; 8 waves x 2 tiles cover the 256 butterflies per stage.
template <int NS>
__device__ __forceinline__ void big_stage(const float2 *Ls, float2 *Ld,
                                          const v2f *ar, const v2f *ai,
                                          float dirSign, int lane, int warp) {
    const int n     = lane & 15;
    const int khalf = (lane >> 4) << 1;
    const int rbase = (lane >> 4) << 3;
#pragma unroll
    for (int tt = 0; tt < 2; ++tt) {
        const int j = (warp * 2 + tt) * 16 + n;
        WMMA_CPLX_TILE(Ls[j + 256 * k0], Ls[j + 256 * (k0 + 1)])
        const int jm = j & (NS - 1);
        const int jd = j / NS;                       // NS compile-time -> shift
        float2 *dp = Ld + jd * (NS * 16) + jm;
        const float wb = dirSign * (2.0f * PI_F) * (float)jm / (float)(NS * 16);
#pragma unroll
        for (int v = 0; v < 8; ++v) {
            const int r = rbase + v;
            float s, cw; __sincosf(wb * (float)r, &s, &cw);
            dp[r * NS] = make_float2(yr[v] * cw - yi[v] * s, yr[v] * s + yi[v] * cw);
        }
    }
}

__global__ __launch_bounds__(256)
void fft4096_wmma(const float2 *__restrict__ src, float2 *__restrict__ dst,
                  float dirSign) {
    __shared__ float2 L0[4096];
    __shared__ float2 L1[4096];
    const int tid  = threadIdx.x;
    const int lane = tid & 31;
    const int warp = tid >> 5;
    const size_t rowBase = (size_t)blockIdx.x * 4096;

    const float4 *s4 = (const float4 *)(src + rowBase);
    float4 *l4 = (float4 *)L0;
#pragma unroll
    for (int i = 0; i < 8; ++i) l4[tid + 256 * i] = s4[tid + 256 * i];

    v2f ar[4], ai[4];
    dft16_frags(lane, dirSign, ar, ai);
    __syncthreads();
    big_stage<1>(L0, L1, ar, ai, dirSign, lane, warp);
    __syncthreads();
    big_stage<16>(L1, L0, ar, ai, dirSign, lane, warp);
    __syncthreads();
    big_stage<256>(L0, L1, ar, ai, dirSign, lane, warp);
    __syncthreads();

    float4 *d4 = (float4 *)(dst + rowBase);
    const float4 *l14 = (const float4 *)L1;
#pragma unroll
    for (int i = 0; i < 8; ++i) d4[tid + 256 * i] = l14[tid + 256 * i];
}

// ================ four-step FFT: 128-pt + N-twiddle LDS kernel =============
// Tile = 32 outer (n1) x 128 inner (FFT axis). Forward: natural -> T[k2][n1]
// with post-twiddle. Inverse: T[k2][n1] -> natural with pre-twiddle.
// (Load/store addressing is identical in both directions.)
__global__ __launch_bounds__(256)
void fft128_twiddle(const float2 *__restrict__ src, float2 *__restrict__ dst,
                    float dirSign, int inverse) {
    __shared__ float2 L0[4096];    // [m][o]: m*32+o
    __shared__ float2 L1[4096];
    const int tid    = threadIdx.x;
    const int lane   = tid & 31;
    const int warp   = tid >> 5;
    const int sig    = blockIdx.x >> 7;            // 128 WGs per signal
    const int n1base = (blockIdx.x & 127) << 5;    // 32 n1 per WG
    const float2 *sp = src + (size_t)sig * NFA;
    float2 *dp = dst + (size_t)sig * NFA;
    const int o    = tid & 31;
    const int mrow = tid >> 5;
    const float twk = dirSign * (2.0f * PI_F) / (float)NFA;

    // ---- load (pre-twiddle when inverse) ----
#pragma unroll 4
    for (int it = 0; it < 16; ++it) {
        int m  = it * 8 + mrow;
        int n1 = n1base + o;
        float2 v = sp[(size_t)m * N1F + n1];
        if (inverse) {
            float s, c; __sincosf(twk * (float)(n1 * m), &s, &c);
            v = make_float2(v.x * c - v.y * s, v.x * s + v.y * c);
        }
        L0[m * 32 + o] = v;
    }
    __syncthreads();

    // ---- stage 1: radix-16 (Ns=1), j in [0,8) per 128-pt FFT, WMMA ----
    v2f ar[4], ai[4];
    dft16_frags(lane, dirSign, ar, ai);
    {
        const int n     = lane & 15;
        const int khalf = (lane >> 4) << 1;
        const int rbase = (lane >> 4) << 3;
#pragma unroll
        for (int tt = 0; tt < 2; ++tt) {
            const int col = (warp * 2 + tt) * 16 + n;   // 0..255
            const int oo  = col >> 3;                   // which of 32 FFTs
            const int j   = col & 7;
            WMMA_CPLX_TILE(L0[(j + 8 * k0) * 32 + oo], L0[(j + 8 * (k0 + 1)) * 32 + oo])
#pragma unroll
            for (int v = 0; v < 8; ++v) {
                const int r = rbase + v;
                L1[(j * 16 + r) * 32 + oo] = make_float2(yr[v], yi[v]);
            }
        }
    }
    __syncthreads();

    // ---- stage 2: radix-8 (Ns=16), VALU ----
#pragma unroll
    for (int it = 0; it < 2; ++it) {
        const int b  = tid + 256 * it;
        const int oo = b >> 4;
        const int j  = b & 15;
        float2 x[8];
#pragma unroll
        for (int k = 0; k < 8; ++k) x[k] = L1[(j + 16 * k) * 32 + oo];
#pragma unroll
        for (int r = 0; r < 8; ++r) {
            float accr = 0.f, acci = 0.f;
#pragma unroll
            for (int k = 0; k < 8; ++k) {
                float ang = dirSign * (PI_F / 4.0f) * (float)((r * k) & 7);
                float s, c; __sincosf(ang, &s, &c);
                accr += x[k].x * c - x[k].y * s;
                acci += x[k].x * s + x[k].y * c;
            }
            float ang = dirSign * (2.0f * PI_F / 128.0f) * (float)(r * j);
            float s, c; __sincosf(ang, &s, &c);
            L0[(r * 16 + j) * 32 + oo] = make_float2(accr * c - acci * s, accr * s + acci * c);
        }
    }
    __syncthreads();

    // ---- store (post-twiddle when forward) ----
#pragma unroll 4
    for (int it = 0; it < 16; ++it) {
        int m  = it * 8 + mrow;
        int n1 = n1base + o;
        float2 v = L0[m * 32 + o];
        if (!inverse) {
            float s, c; __sincosf(twk * (float)(n1 * m), &s, &c);
            v = make_float2(v.x * c - v.y * s, v.x * s + v.y * c);
        }
        dp[(size_t)m * N1F + n1] = v;
    }
}

// ================ global Stockham stages (N = 2^17 IR path) ================
__global__ __launch_bounds__(256)
void fft_r16_wmma(const float2 *__restrict__ src, float2 *__restrict__ dst,
                  int logN, int logNs, float dirSign) {
    const int wid   = (int)((blockIdx.x * (size_t)blockDim.x + threadIdx.x) >> 5);
    const int lane  = threadIdx.x & 31;
    const int lwps  = logN - 8;                 // log2(wavesPerSig)
    const int sig   = wid >> lwps;
    const int j0    = (wid & ((1 << lwps) - 1)) << 4;
    const int RS    = 1 << (logN - 4);
    const int Ns    = 1 << logNs;
    const int khalf = (lane >> 4) << 1;
    const int rbase = (lane >> 4) << 3;

    v2f ar[4], ai[4];
    dft16_frags(lane, dirSign, ar, ai);

    const float2 *sp = src + ((size_t)sig << logN);
    const int j = j0 + (lane & 15);

    WMMA_CPLX_TILE(sp[(size_t)j + (size_t)RS * k0], sp[(size_t)j + (size_t)RS * (k0 + 1)])

    const int jm = j & (Ns - 1);
    const int jd = j >> logNs;
    float2 *dp = dst + ((size_t)sig << logN) + (size_t)jd * ((size_t)Ns << 4) + jm;
    const float wb = dirSign * (2.0f * PI_F) * (float)jm / (float)(Ns << 4);
#pragma unroll
    for (int v = 0; v < 8; ++v) {
        const int r = rbase + v;
        float s, cw; __sincosf(wb * (float)r, &s, &cw);
        dp[(size_t)r * Ns] = make_float2(yr[v] * cw - yi[v] * s, yr[v] * s + yi[v] * cw);
    }
}

template <int R, int LR>
__global__ void fft_small(const float2 *__restrict__ src, float2 *__restrict__ dst,
                          int logN, int logNs, float dirSign, int total) {
    int tid = (int)(blockIdx.x * (size_t)blockDim.x + threadIdx.x);
    if (tid >= total) return;
    const int logNR = logN - LR;
    const int sig = tid >> logNR;
    const int j   = tid & ((1 << logNR) - 1);
    const int NR  = 1 << logNR;
    const int Ns  = 1 << logNs;
    const float2 *sp = src + ((size_t)sig << logN);
    float2 x[R];
#pragma unroll
    for (int k = 0; k < R; ++k) x[k] = sp[(size_t)j + (size_t)NR * k];
    const int jm = j & (Ns - 1);
    const int jd = j >> logNs;
    float2 *dp = dst + ((size_t)sig << logN) + (size_t)jd * ((size_t)Ns * R) + jm;
    const float wtw = dirSign * 2.0f * PI_F * (float)jm / (float)(Ns * R);
#pragma unroll
    for (int r = 0; r < R; ++r) {
        float accr = 0.f, acci = 0.f;
#pragma unroll
        for (int k = 0; k < R; ++k) {
            float ang = dirSign * (2.0f * PI_F / (float)R) * (float)((r * k) & (R - 1));
            float s, c; __sincosf(ang, &s, &c);
            accr += x[k].x * c - x[k].y * s;
            acci += x[k].x * s + x[k].y * c;
        }
        float s, c; __sincosf(wtw * (float)r, &s, &c);
        dp[(size_t)r * Ns] = make_float2(accr * c - acci * s, accr * s + acci * c);
    }
}

// ------------------------------ pipeline kernels ---------------------------
__global__ void k_prep(const float *__restrict__ tp, const float *__restrict__ mp,
                       float *__restrict__ P) {
    int i = (int)(blockIdx.x * blockDim.x + threadIdx.x);
    if (i < BSZ * NTR * 27) {
        int c = i % 27;
        P[i] = tp[i] * (TRACK_R[c][1] - TRACK_R[c][0]) + TRACK_R[c][0];
    } else if (i < BSZ * NTR * 27 + BSZ * 26) {
        int jj = i - BSZ * NTR * 27;
        int c = jj % 26;
        P[BSZ * NTR * 27 + jj] = mp[jj] * (MASTER_R[c][1] - MASTER_R[c][0]) + MASTER_R[c][0];
    }
}

__global__ void k_pack_tracks(float2 *__restrict__ A, const float *__restrict__ tracks,
                              const float *__restrict__ P) {
    size_t idx = blockIdx.x * (size_t)blockDim.x + threadIdx.x;  // 32*NFA
    int t   = (int)(idx & (size_t)(NFA - 1));
    int sig = (int)(idx >> LOGA);
    float v = 0.f;
    if (t < SEQL) {
        float g = __expf(P[sig * 27] * DB2LIN);
        const float *src = tracks + (size_t)sig * SEQL;
        v = src[t] * g;
        if (t + 4096 < SEQL) __builtin_prefetch(&src[t + 4096], 0, 1);  // global_prefetch_b8
    }
    A[idx] = make_float2(v, 0.f);
}

// storage index -> frequency bin (four-step scrambled layout when flag set)
__device__ __forceinline__ int bin_of(int t, int scrambled) {
    if (!scrambled) return t;
    int k2 = t >> 12;           // / N1F
    int k1 = t & (N1F - 1);
    return k2 + (k1 << 7);      // k2 + N2F*k1
}

__global__ void k_eq_mul(float2 *__restrict__ X, const float *__restrict__ P,
                         int pstride, int isMaster, int logN, int scrambled) {
    size_t idx = blockIdx.x * (size_t)blockDim.x + threadIdx.x;
    const int N = 1 << logN;
    int t   = (int)(idx & (size_t)(N - 1));
    int sig = (int)(idx >> logN);
    int row = isMaster ? (sig >> 1) : sig;
    const float *p = P + (size_t)row * pstride + 1;
    float w = (2.f * PI_F / (float)N) * (float)bin_of(t, scrambled);
    float sw, cw; __sincosf(w, &sw, &cw);
    float2 z1 = make_float2(cw, -sw);
    float2 z2 = cmul(z1, z1);
    float2 H = make_float2(1.f, 0.f);
    H = cmul(H, biquad_h(p[0],  p[1],  p[2],  0, z1, z2));
    H = cmul(H, biquad_h(p[3],  p[4],  p[5],  1, z1, z2));
    H = cmul(H, biquad_h(p[6],  p[7],  p[8],  1, z1, z2));
    H = cmul(H, biquad_h(p[9],  p[10], p[11], 1, z1, z2));
    H = cmul(H, biquad_h(p[12], p[13], p[14], 1, z1, z2));
    H = cmul(H, biquad_h(p[15], p[16], p[17], 2, z1, z2));
    X[idx] = cmul(X[idx], H);
}

__global__ void k_comp_gain(const float2 *__restrict__ xt, float2 *__restrict__ g,
                            const float *__restrict__ P, int pstride, int isMaster,
                            int logN, float scaleX) {
    size_t idx = blockIdx.x * (size_t)blockDim.x + threadIdx.x;
    const int N = 1 << logN;
    int t   = (int)(idx & (size_t)(N - 1));
    int sig = (int)(idx >> logN);
    float out = 0.f;
    if (t < SEQL) {
        int row = isMaster ? (sig >> 1) : sig;
        const float *p = P + (size_t)row * pstride + 19;
        float x = xt[idx].x * scaleX;
        float xdb = 20.f * __log10f(fmaxf(fabsf(x), 1e-8f));
        float th = p[0], ratio = p[1], knee = p[4];
        float d = xdb - th;
        float xsc;
        if (2.f * d < -knee) xsc = xdb;
        else if (2.f * fabsf(d) <= knee) {
            float e = d + knee * 0.5f;
            xsc = xdb + (1.f / ratio - 1.f) * e * e / (2.f * knee);
        } else xsc = th + d / ratio;
        out = xsc - xdb;
    }
    g[idx] = make_float2(out, 0.f);
}

__global__ void k_comp_smooth_mul(float2 *__restrict__ G, const float *__restrict__ P,
                                  int pstride, int isMaster, int logN, int scrambled) {
    size_t idx = blockIdx.x * (size_t)blockDim.x + threadIdx.x;
    const int N = 1 << logN;
    int t   = (int)(idx & (size_t)(N - 1));
    int sig = (int)(idx >> logN);
    int row = isMaster ? (sig >> 1) : sig;
    const float *p = P + (size_t)row * pstride + 19;
    float at = p[2];
    float alpha = __expf(-2.19722458f / (44.1f * at));
    float w = (2.f * PI_F / (float)N) * (float)bin_of(t, scrambled);
    float sw, cw; __sincosf(w, &sw, &cw);
    float2 den = make_float2(1.f - alpha * cw, alpha * sw);
    float2 num = make_float2(1.f - alpha, 0.f);
    G[idx] = cmul(G[idx], cdiv(num, den));
}

__global__ void k_comp_apply(float2 *__restrict__ xt, const float2 *__restrict__ gt,
                             const float *__restrict__ P, int pstride, int isMaster,
                             int logN, int LA, float scaleX, float scaleG) {
    size_t idx = blockIdx.x * (size_t)blockDim.x + threadIdx.x;  // batch*SEQ
    const int N = 1 << logN;
    int t   = (int)(idx % (size_t)SEQL);
    int sig = (int)(idx / (size_t)SEQL);
    int row = isMaster ? (sig >> 1) : sig;
    const float *p = P + (size_t)row * pstride + 19;
    float makeup = p[5];
    const float2 *gs = gt + (size_t)sig * N;
    float g = ((t + LA) < SEQL ? gs[t + LA].x : gs[SEQL - 1].x) * scaleG;
    size_t xi = (size_t)sig * N + t;
    float y = xt[xi].x * scaleX * __expf((g + makeup) * DB2LIN);
    xt[xi] = make_float2(y, 0.f);
}

__global__ void k_pan_sum(const float2 *__restrict__ A, const float *__restrict__ P,
                          float *__restrict__ panned, float *__restrict__ masterBuf,
                          float *__restrict__ fxinBuf) {
    size_t idx = blockIdx.x * (size_t)blockDim.x + threadIdx.x;  // BSZ*2*SEQ
    int t  = (int)(idx % (size_t)SEQL);
    int bc = (int)(idx / (size_t)SEQL);
    int ch = bc & 1, b = bc >> 1;
    float msum = 0.f, fsum = 0.f;
    for (int tr = 0; tr < NTR; ++tr) {
        int sig = b * NTR + tr;
        const float *p = P + sig * 27;
        float theta = p[26] * (PI_F * 0.5f);
        float st, ct; __sincosf(theta, &st, &ct);
        float g = (ch == 0)
                      ? sqrtf(fmaxf((PI_F * 0.5f - theta) * (2.0f / PI_F) * ct, 0.f))
                      : sqrtf(fmaxf(theta * (2.0f / PI_F) * st, 0.f));
        float y = A[((size_t)sig << LOGA) + t].x;
        float pv = y * g;
        panned[((size_t)(b * 2 + ch) * NTR + tr) * SEQL + t] = pv;
        msum += pv;
        fsum += pv * __expf(p[25] * DB2LIN);
    }
    masterBuf[(size_t)bc * SEQL + t] = msum;
    fxinBuf[(size_t)bc * SEQL + t]  = fsum;
}

__global__ void k_pack_noise(float2 *__restrict__ A, const float *__restrict__ noise) {
    size_t idx = blockIdx.x * (size_t)blockDim.x + threadIdx.x;  // 48*NFB
    int t   = (int)(idx & (size_t)(NFB - 1));
    int sig = (int)(idx >> LOGB);
    float v = (t < NOISEL) ? noise[(size_t)sig * NOISEL + t] : 0.f;
    A[idx] = make_float2(v, 0.f);
}

__global__ void k_pack_fb(float2 *__restrict__ C, const float *__restrict__ fb) {
    size_t idx = blockIdx.x * (size_t)blockDim.x + threadIdx.x;  // 12*NFB
    int t = (int)(idx & (size_t)(NFB - 1));
    int f = (int)(idx >> LOGB);
    float v = (t < TAPSN) ? fb[(size_t)f * TAPSN + t] : 0.f;
    C[idx] = make_float2(v, 0.f);
}

__global__ void k_cmul_rows(float2 *__restrict__ X, const float2 *__restrict__ Y,
                            int logN, int ymod) {
    size_t idx = blockIdx.x * (size_t)blockDim.x + threadIdx.x;
    const int N = 1 << logN;
    int t   = (int)(idx & (size_t)(N - 1));
    int sig = (int)(idx >> logN);
    X[idx] = cmul(X[idx], Y[((size_t)(sig % ymod) << logN) + t]);
}

__global__ void k_ir_build(const float2 *__restrict__ filt, const float *__restrict__ fxp,
                           float *__restrict__ irBuf) {
    size_t idx = blockIdx.x * (size_t)blockDim.x + threadIdx.x;  // 4*REVN
    int t   = (int)(idx & (size_t)(REVN - 1));
    int sig = (int)(idx >> 16);
    int b = sig >> 1;
    float tl = (float)t / (float)(REVN - 1);
    float acc = 0.f;
    for (int f = 0; f < 12; ++f) {
        float dec = fxp[b * 25 + 12 + f] * 10.f + 1.f;
        float gn  = fxp[b * 25 + f];
        float v = filt[((size_t)(sig * 12 + f) << LOGB) + (TAPSN - 1) + t].x * (1.f / (float)NFB);
        acc += v * __expf(-dec * tl) * gn;
    }
    irBuf[idx] = acc * (1.f / 12.f);
}

__global__ void k_pack_real(float2 *__restrict__ A, const float *__restrict__ src,
                            int logN, int len, float gain) {
    size_t idx = blockIdx.x * (size_t)blockDim.x + threadIdx.x;  // batch*N
    const int N = 1 << logN;
    int t   = (int)(idx & (size_t)(N - 1));
    int sig = (int)(idx >> logN);
    float v = (t < len) ? src[(size_t)sig * len + t] * gain : 0.f;
    A[idx] = make_float2(v, 0.f);
}

__global__ void k_pack_master(float2 *__restrict__ A, const float *__restrict__ masterBuf,
                              const float *__restrict__ PM) {
    size_t idx = blockIdx.x * (size_t)blockDim.x + threadIdx.x;  // 4*NFA
    int t   = (int)(idx & (size_t)(NFA - 1));
    int sig = (int)(idx >> LOGA);
    int b = sig >> 1;
    float v = 0.f;
    if (t < SEQL) v = masterBuf[(size_t)sig * SEQL + t] * __expf(PM[b * 26] * DB2LIN);
    A[idx] = make_float2(v, 0.f);
}

__global__ void k_wet_mix(float *__restrict__ masterBuf, const float *__restrict__ fxinBuf,
                          const float2 *__restrict__ wet, const float *__restrict__ fxp,
                          float scale) {
    size_t idx = blockIdx.x * (size_t)blockDim.x + threadIdx.x;  // 4*SEQ
    int t   = (int)(idx % (size_t)SEQL);
    int sig = (int)(idx / (size_t)SEQL);
    int b = sig >> 1;
    float m = fxp[b * 25 + 24];
    float w = wet[((size_t)sig << LOGA) + t].x * scale;
    masterBuf[idx] = masterBuf[idx] + (1.f - m) * fxinBuf[idx] + m * w;
}

__global__ void k_master_out(float *__restrict__ out, const float2 *__restrict__ xt,
                             const float *__restrict__ PM) {
    size_t idx = blockIdx.x * (size_t)blockDim.x + threadIdx.x;  // 4*SEQ
    int t   = (int)(idx % (size_t)SEQL);
    int sig = (int)(idx / (size_t)SEQL);
    int b = sig >> 1;
    out[idx] = xt[((size_t)sig << LOGA) + t].x * __expf(PM[b * 26 + 25] * DB2LIN);
}

// ------------------------------- host helpers ------------------------------
static inline unsigned blocksFor(size_t total) { return (unsigned)((total + 255) / 256); }

// Four-step FFT at N = 2^19; result lands back in x (scratch is clobbered).
static void run_big(float2 *x, float2 *scratch, int batch, float dirSign,
                    hipStream_t stream) {
    const unsigned wgs = (unsigned)batch * 128;
    if (dirSign < 0.f) {  // forward: small(+post twiddle) then big
        fft128_twiddle<<<wgs, 256, 0, stream>>>(x, scratch, dirSign, 0);
        fft4096_wmma<<<wgs, 256, 0, stream>>>(scratch, x, dirSign);
    } else {              // inverse: big then small(+pre twiddle)
        fft4096_wmma<<<wgs, 256, 0, stream>>>(x, scratch, dirSign);
        fft128_twiddle<<<wgs, 256, 0, stream>>>(scratch, x, dirSign, 1);
    }
}

// Global mixed-radix Stockham (used for N = 2^17); returns result buffer.
static float2 *run_fft(float2 *x, float2 *tmp, int batch, int logN, float dirSign,
                       hipStream_t stream) {
    float2 *s = x, *d = tmp;
    int logNs = 0, logRem = logN;
    while (logRem > 0) {
        if (logRem >= 4) {
            size_t threads = ((size_t)batch << logN) >> 3;  // batch*N/16 butterflies *32/16... = batch*N/8 lanes
            fft_r16_wmma<<<blocksFor(threads), 256, 0, stream>>>(s, d, logN, logNs, dirSign);
            logNs += 4; logRem -= 4;
        } else if (logRem == 3) {
            int total = (batch << logN) >> 3;
            fft_small<8, 3><<<blocksFor((size_t)total), 256, 0, stream>>>(s, d, logN, logNs, dirSign, total);
            logNs += 3; logRem -= 3;
        } else if (logRem == 2) {
            int total = (batch << logN) >> 2;
            fft_small<4, 2><<<blocksFor((size_t)total), 256, 0, stream>>>(s, d, logN, logNs, dirSign, total);
            logNs += 2; logRem -= 2;
        } else {
            int total = (batch << logN) >> 1;
            fft_small<2, 1><<<blocksFor((size_t)total), 256, 0, stream>>>(s, d, logN, logNs, dirSign, total);
            logNs += 1; logRem -= 1;
        }
        float2 *t2 = s; s = d; d = t2;
    }
    return s;
}

extern "C" void kernel_launch(void *const *d_in, const int *in_sizes, int n_in,
                              void *d_out, int out_size, void *d_ws, size_t ws_size,
                              hipStream_t stream) {
    (void)in_sizes; (void)n_in; (void)out_size; (void)ws_size;
    const float *tracks = (const float *)d_in[0];
    const float *tparam = (const float *)d_in[1];
    const float *fxp    = (const float *)d_in[2];
    const float *mparam = (const float *)d_in[3];
    const float *noise  = (const float *)d_in[4];
    const float *fb     = (const float *)d_in[5];
    float *out = (float *)d_out;

    const size_t CAPC = (size_t)BSZ * NTR * NFA;   // 32 * 2^19 complex
    float2 *A  = (float2 *)d_ws;
    float2 *Bb = A + CAPC;
    float2 *Cc = Bb + CAPC;
    float *P        = (float *)(Cc + CAPC);
    float *PM       = P + BSZ * NTR * 27;
    float *masterBf = P + 1024;
    float *fxinBf   = masterBf + (size_t)BSZ * 2 * SEQL;
    float *irBf     = fxinBf + (size_t)BSZ * 2 * SEQL;

    const float invA = 1.f / (float)NFA;
    const int NSIG = BSZ * NTR;

    k_prep<<<1, 1024, 0, stream>>>(tparam, mparam, P);

    // ---- tracks: gain -> EQ -> compressor ----
    k_pack_tracks<<<blocksFor((size_t)NSIG * NFA), 256, 0, stream>>>(A, tracks, P);
    run_big(A, Bb, NSIG, -1.f, stream);
    k_eq_mul<<<blocksFor((size_t)NSIG * NFA), 256, 0, stream>>>(A, P, 27, 0, LOGA, 1);
    run_big(A, Bb, NSIG, +1.f, stream);
    k_comp_gain<<<blocksFor((size_t)NSIG * NFA), 256, 0, stream>>>(A, Cc, P, 27, 0, LOGA, invA);
    run_big(Cc, Bb, NSIG, -1.f, stream);
    k_comp_smooth_mul<<<blocksFor((size_t)NSIG * NFA), 256, 0, stream>>>(Cc, P, 27, 0, LOGA, 1);
    run_big(Cc, Bb, NSIG, +1.f, stream);
    k_comp_apply<<<blocksFor((size_t)NSIG * SEQL), 256, 0, stream>>>(A, Cc, P, 27, 0, LOGA,
                                                                    2048, invA, invA);
    k_pan_sum<<<blocksFor((size_t)BSZ * 2 * SEQL), 256, 0, stream>>>(A, P, out, masterBf, fxinBf);

    // ---- reverb IR build (2^17 Stockham path) ----
    k_pack_fb<<<blocksFor((size_t)12 * NFB), 256, 0, stream>>>(Cc, fb);
    float2 *FBs = run_fft(Cc, Bb, 12, LOGB, -1.f, stream);
    k_pack_noise<<<blocksFor((size_t)48 * NFB), 256, 0, stream>>>(A, noise);
    float2 *NSp = run_fft(A, (FBs == Bb) ? Cc : Bb, 48, LOGB, -1.f, stream);
    k_cmul_rows<<<blocksFor((size_t)48 * NFB), 256, 0, stream>>>(NSp, FBs, LOGB, 12);
    float2 *Ft = run_fft(NSp, (NSp == Cc) ? A : Cc, 48, LOGB, +1.f, stream);
    k_ir_build<<<blocksFor((size_t)4 * REVN), 256, 0, stream>>>(Ft, fxp, irBf);

    // ---- wet = fx_in (*) ir ----
    k_pack_real<<<blocksFor((size_t)4 * NFA), 256, 0, stream>>>(A, fxinBf, LOGA, SEQL, 1.f);
    k_pack_real<<<blocksFor((size_t)4 * NFA), 256, 0, stream>>>(Cc, irBf, LOGA, REVN, 1.f);
    run_big(Cc, Bb, 4, -1.f, stream);
    run_big(A, Bb, 4, -1.f, stream);
    k_cmul_rows<<<blocksFor((size_t)4 * NFA), 256, 0, stream>>>(A, Cc, LOGA, 4);
    run_big(A, Bb, 4, +1.f, stream);
    k_wet_mix<<<blocksFor((size_t)4 * SEQL), 256, 0, stream>>>(masterBf, fxinBf, A, fxp, invA);

    // ---- master chain ----
    k_pack_master<<<blocksFor((size_t)4 * NFA), 256, 0, stream>>>(A, masterBf, PM);
    run_big(A, Bb, 4, -1.f, stream);
    k_eq_mul<<<blocksFor((size_t)4 * NFA), 256, 0, stream>>>(A, PM, 26, 1, LOGA, 1);
    run_big(A, Bb, 4, +1.f, stream);
    k_comp_gain<<<blocksFor((size_t)4 * NFA), 256, 0, stream>>>(A, Cc, PM, 26, 1, LOGA, invA);
    run_big(Cc, Bb, 4, -1.f, stream);
    k_comp_smooth_mul<<<blocksFor((size_t)4 * NFA), 256, 0, stream>>>(Cc, PM, 26, 1, LOGA, 1);
    run_big(Cc, Bb, 4, +1.f, stream);
    k_comp_apply<<<blocksFor((size_t)4 * SEQL), 256, 0, stream>>>(A, Cc, PM, 26, 1, LOGA,
                                                                 1024, invA, invA);
    k_master_out<<<blocksFor((size_t)4 * SEQL), 256, 0, stream>>>(out + PANNED_SZ, A, PM);
}